// GClassifier_3521873183176
// MI455X (gfx1250) — compile-verified
//
#include <hip/hip_runtime.h>
#include <hip/hip_bf16.h>

typedef __attribute__((ext_vector_type(2))) float v2f;
typedef __attribute__((ext_vector_type(8))) float v8f;

#define N_NODES 50000
#define N_EDGES 800000
#define DFEAT 128
#define N_CLASSES 10
#define SA_STRIDE 132  // padded LDS row stride: bank(4m+ka) -> conflict-free

// ---------------------------------------------------------------------------
// Degree / normalization
// ---------------------------------------------------------------------------
__global__ void gcn_init_deg(float* __restrict__ deg) {
    int i = blockIdx.x * blockDim.x + threadIdx.x;
    if (i < N_NODES) deg[i] = 1.0f;  // self-loop weight
}

__global__ void gcn_accum_deg(const int* __restrict__ ei,
                              const float* __restrict__ ew,
                              float* __restrict__ deg) {
    int e = blockIdx.x * blockDim.x + threadIdx.x;
    if (e < N_EDGES) {
        int c = ei[N_EDGES + e];  // col = target
        atomicAdd(&deg[c], ew[e]);
    }
}

__global__ void gcn_make_dinv(float* __restrict__ deg) {
    int i = blockIdx.x * blockDim.x + threadIdx.x;
    if (i < N_NODES) {
        float d = deg[i];
        deg[i] = (d > 0.0f) ? rsqrtf(d) : 0.0f;
    }
}

// ---------------------------------------------------------------------------
// Dense GEMM  Y[N,128] = X[N,128] @ W[128,128]  via V_WMMA_F32_16X16X4_F32
// grid.x = N/16 blocks, 256 threads = 8 waves; wave w owns output cols
// [16w, 16w+16). The 16x128 A tile is cooperatively staged into LDS with
// coalesced float4 loads; fragments then come from LDS (conflict-free via
// 132-float padded stride). W fragment loads are naturally coalesced and
// L2-resident. 32 K-steps of K=4 per wave; EXEC stays all-1s (no divergence).
// ---------------------------------------------------------------------------
__global__ void gcn_gemm_wmma(const float* __restrict__ X,
                              const float* __restrict__ W,
                              float* __restrict__ Y) {
    __shared__ float sA[16 * SA_STRIDE];

    const int tid  = threadIdx.x;
    const int lane = tid & 31;
    const int wave = tid >> 5;
    const int row0 = blockIdx.x << 4;  // 16 rows per block

    // cooperative A-tile stage: 16 threads per row, 2 float4 per thread
    {
        const int r  = tid >> 4;
        const int c4 = tid & 15;
        const float4* __restrict__ src =
            (const float4*)(X + (size_t)(row0 + r) * DFEAT);
        float4* dst = (float4*)(sA + r * SA_STRIDE);
        dst[c4]      = src[c4];
        dst[c4 + 16] = src[c4 + 16];
    }
    __syncthreads();

    const int col0 = wave << 4;   // 16 cols per wave
    const int m    = lane & 15;   // A row / B col / D col index
    const int hi   = lane >> 4;   // K-half selector (0 or 1)

    const float* __restrict__ sArow = sA + m * SA_STRIDE;

    v8f acc = {};
    #pragma unroll 4
    for (int k = 0; k < DFEAT; k += 4) {
        const int ka = k + (hi << 1);
        v2f a, b;
        // A 16x4 fragment from LDS: lane m holds A[m][ka], A[m][ka+1]
        a.x = sArow[ka];
        a.y = sArow[ka + 1];
        // B 4x16 fragment: lane m holds W[ka][col0+m], W[ka+1][col0+m]
        b.x = W[(size_t)ka * DFEAT + col0 + m];
        b.y = W[(size_t)(ka + 1) * DFEAT + col0 + m];
        acc = __builtin_amdgcn_wmma_f32_16x16x4_f32(
            /*neg_a=*/false, a, /*neg_b=*/false, b,
            /*c_mod=*/(short)0, acc, /*reuse_a=*/false, /*reuse_b=*/false);
    }

    // D 16x16: VGPR v holds row (8*hi + v), col m
    const int mbase = hi << 3;
    #pragma unroll
    for (int v = 0; v < 8; ++v) {
        Y[(size_t)(row0 + mbase + v) * DFEAT + col0 + m] = acc[v];
    }
}

// ---------------------------------------------------------------------------
// Aggregation: self-loop term (init), then edge-wise gather-scale-scatter
// ---------------------------------------------------------------------------
__global__ void gcn_agg_self(const float* __restrict__ dinv,
                             const float* __restrict__ xw,
                             float* __restrict__ agg) {
    size_t idx = (size_t)blockIdx.x * blockDim.x + threadIdx.x;
    if (idx < (size_t)N_NODES * DFEAT) {
        int i = (int)(idx >> 7);  // DFEAT == 128
        float s = dinv[i];
        agg[idx] = s * s * xw[idx];
    }
}

__global__ void gcn_agg_edges(const int* __restrict__ ei,
                              const float* __restrict__ ew,
                              const float* __restrict__ dinv,
                              const float* __restrict__ xw,
                              float* __restrict__ agg) {
    // one wave32 per edge; each lane owns 4 consecutive floats (128/32)
    int wid  = (int)(((size_t)blockIdx.x * blockDim.x + threadIdx.x) >> 5);
    int lane = threadIdx.x & 31;
    if (wid >= N_EDGES) return;
    int r = ei[wid];
    int c = ei[N_EDGES + wid];
    float norm = dinv[r] * ew[wid] * dinv[c];
    const float4* __restrict__ src =
        (const float4*)(xw + (size_t)r * DFEAT);
    float* __restrict__ dst = agg + (size_t)c * DFEAT;
    float4 v = src[lane];
    int d = lane << 2;
    atomicAdd(&dst[d + 0], norm * v.x);
    atomicAdd(&dst[d + 1], norm * v.y);
    atomicAdd(&dst[d + 2], norm * v.z);
    atomicAdd(&dst[d + 3], norm * v.w);
}

__global__ void gcn_bias_relu(float* __restrict__ h, const float* __restrict__ b) {
    size_t idx = (size_t)blockIdx.x * blockDim.x + threadIdx.x;
    if (idx < (size_t)N_NODES * DFEAT) {
        int d = (int)(idx & (DFEAT - 1));
        float v = h[idx] + b[d];
        h[idx] = v > 0.0f ? v : 0.0f;
    }
}

// ---------------------------------------------------------------------------
// Readout: column sums then tiny matvec
// ---------------------------------------------------------------------------
__global__ void gcn_zero_gsum(float* __restrict__ gsum) {
    gsum[threadIdx.x] = 0.0f;
}

__global__ void gcn_colsum(const float* __restrict__ h, float* __restrict__ gsum) {
    int d = threadIdx.x;  // 128 threads = one column each
    float s = 0.0f;
    for (int i = blockIdx.x; i < N_NODES; i += gridDim.x) {
        s += h[(size_t)i * DFEAT + d];
    }
    atomicAdd(&gsum[d], s);
}

__global__ void gcn_final(const float* __restrict__ gsum,
                          const float* __restrict__ Wm,
                          const float* __restrict__ bm,
                          float* __restrict__ out) {
    int c = threadIdx.x;
    if (c < N_CLASSES) {
        const float invn = 1.0f / (float)N_NODES;
        float acc = bm[c];
        #pragma unroll 4
        for (int d = 0; d < DFEAT; ++d) {
            acc += (gsum[d] * invn) * Wm[d * N_CLASSES + c];
        }
        out[c] = acc;
    }
}

// ---------------------------------------------------------------------------
extern "C" void kernel_launch(void* const* d_in, const int* in_sizes, int n_in,
                              void* d_out, int out_size, void* d_ws, size_t ws_size,
                              hipStream_t stream) {
    const float* x   = (const float*)d_in[0];
    const int*   ei  = (const int*)d_in[1];   // [2, E] row-major: row then col
    const float* ew  = (const float*)d_in[2];
    const float* W1  = (const float*)d_in[3];
    const float* b1  = (const float*)d_in[4];
    const float* W2  = (const float*)d_in[5];
    const float* b2  = (const float*)d_in[6];
    const float* Wm  = (const float*)d_in[7];
    const float* bm  = (const float*)d_in[8];
    float* out = (float*)d_out;

    float* ws   = (float*)d_ws;
    float* deg  = ws;                                   // N_NODES (-> dinv in place)
    float* xw   = ws + ((N_NODES + 255) & ~255);        // N_NODES*128
    float* agg  = xw + (size_t)N_NODES * DFEAT;         // N_NODES*128
    float* gsum = agg + (size_t)N_NODES * DFEAT;        // 128

    const int nd_blocks = (int)(((size_t)N_NODES * DFEAT + 255) / 256);

    // normalization coefficients
    gcn_init_deg<<<(N_NODES + 255) / 256, 256, 0, stream>>>(deg);
    gcn_accum_deg<<<(N_EDGES + 255) / 256, 256, 0, stream>>>(ei, ew, deg);
    gcn_make_dinv<<<(N_NODES + 255) / 256, 256, 0, stream>>>(deg);

    // ---- layer 1 ----
    gcn_gemm_wmma<<<N_NODES / 16, 256, 0, stream>>>(x, W1, xw);
    gcn_agg_self<<<nd_blocks, 256, 0, stream>>>(deg, xw, agg);
    gcn_agg_edges<<<(N_EDGES * 32 + 255) / 256, 256, 0, stream>>>(ei, ew, deg, xw, agg);
    gcn_bias_relu<<<nd_blocks, 256, 0, stream>>>(agg, b1);

    // ---- layer 2 ---- (reads agg as input, writes xw, re-aggregates into agg)
    gcn_gemm_wmma<<<N_NODES / 16, 256, 0, stream>>>(agg, W2, xw);
    gcn_agg_self<<<nd_blocks, 256, 0, stream>>>(deg, xw, agg);
    gcn_agg_edges<<<(N_EDGES * 32 + 255) / 256, 256, 0, stream>>>(ei, ew, deg, xw, agg);
    gcn_bias_relu<<<nd_blocks, 256, 0, stream>>>(agg, b2);

    // ---- readout ----
    gcn_zero_gsum<<<1, DFEAT, 0, stream>>>(gsum);
    gcn_colsum<<<256, DFEAT, 0, stream>>>(agg, gsum);
    gcn_final<<<1, 32, 0, stream>>>(gsum, Wm, bm, out);
}